// aggregator_9216999817565
// MI455X (gfx1250) — compile-verified
//
#include <hip/hip_runtime.h>

typedef __attribute__((ext_vector_type(2))) float v2f;
typedef __attribute__((ext_vector_type(4))) float v4f;
typedef __attribute__((ext_vector_type(8))) float v8f;

#define D_FEAT 32

// ---------------------------------------------------------------------------
// Kernel 1: initialize accumulators.  sum=0, min=+INF, max=-INF, deg=0.
// ---------------------------------------------------------------------------
__global__ void seg_init_kernel(float* __restrict__ sum, float* __restrict__ mn,
                                float* __restrict__ mx, float* __restrict__ deg,
                                int nd, int n_nodes) {
  int i = blockIdx.x * blockDim.x + threadIdx.x;
  if (i < nd) {
    sum[i] = 0.0f;
    mn[i]  =  __builtin_inff();
    mx[i]  = -__builtin_inff();
  }
  if (i < n_nodes) deg[i] = 0.0f;
}

// ---------------------------------------------------------------------------
// Kernel 2: edge scatter.  One thread per (edge, group-of-4-features).
// float4 (B128) load of m, then f32 atomics into L2-resident accumulators.
// ---------------------------------------------------------------------------
__global__ void seg_edge_kernel(const float* __restrict__ m,
                                const int*   __restrict__ dst,
                                float* __restrict__ sum, float* __restrict__ mn,
                                float* __restrict__ mx,  float* __restrict__ deg,
                                int n_edges) {
  long idx   = (long)blockIdx.x * blockDim.x + threadIdx.x;
  long total = (long)n_edges * 8;          // 8 groups of 4 features per edge
  if (idx >= total) return;

  int e = (int)(idx >> 3);
  int g = (int)(idx & 7);

  // Prefetch the m stream ~16 edges (2KB) ahead of this thread's row.
  __builtin_prefetch(m + (long)e * D_FEAT + 16 * D_FEAT, 0, 0);

  int node = dst[e];
  v4f v = *(const v4f*)(m + (long)e * D_FEAT + g * 4);

  long base = (long)node * D_FEAT + g * 4;
  float* sp = sum + base;
  float* np = mn  + base;
  float* xp = mx  + base;

#pragma unroll
  for (int j = 0; j < 4; ++j) {
    atomicAdd(sp + j, v[j]);   // global_atomic_add_f32
    atomicMin(np + j, v[j]);   // global_atomic_min_num_f32
    atomicMax(xp + j, v[j]);   // global_atomic_max_num_f32
  }
  if (g == 0) atomicAdd(deg + node, 1.0f);
}

// ---------------------------------------------------------------------------
// Kernel 3: finalize via V_WMMA_F32_16X16X4_F32.
//   Per 16-element tile:  A[i][k] = {sum, min', max', mean} of element i
//   (A 16x4 f32 layout: lanes 0-15 hold K0/K1 in VGPR0/1, lanes 16-31 K2/K3).
//   B[k][j] = w[k] for all j (columns identical, so only the K slot mapping
//   matters: lanes 0-15 = (w0,w1), lanes 16-31 = (w2,w3)).
//   D[i][j] = out[i] for all j; rows 0-7 live in lanes 0-15 (c[0..7]),
//   rows 8-15 in lanes 16-31.
// ---------------------------------------------------------------------------
__global__ void seg_finalize_wmma_kernel(const float* __restrict__ sum,
                                         const float* __restrict__ mn,
                                         const float* __restrict__ mx,
                                         const float* __restrict__ deg,
                                         const float* __restrict__ w,
                                         const float* __restrict__ bptr,
                                         float* __restrict__ out, int nd) {
  const float w0 = w[0], w1 = w[1], w2 = w[2], w3 = w[3];
  const float bb = bptr[0];

  int lane   = threadIdx.x & 31;
  int wave   = (int)((blockIdx.x * blockDim.x + threadIdx.x) >> 5);
  int nwaves = (int)((gridDim.x * blockDim.x) >> 5);
  int ntiles = nd >> 4;

  bool hi = (lane >= 16);
  int  r  = lane & 15;

  for (int t = wave; t < ntiles; t += nwaves) {
    int base = t << 4;
    int e    = base + r;               // both lane halves load the same element

    float s   = sum[e];
    float mnv = mn[e];
    float mxv = mx[e];
    float dg  = deg[e >> 5];           // D_FEAT == 32
    bool  has = dg > 0.0f;
    mnv = has ? mnv : 0.0f;            // DGL semantics: 0 for zero-degree nodes
    mxv = has ? mxv : 0.0f;
    float mean = s / fmaxf(dg, 1.0f);

    v2f a, bv;
    a.x  = hi ? mxv : s;               // K2 : K0
    a.y  = hi ? mean : mnv;            // K3 : K1
    bv.x = hi ? w2 : w0;
    bv.y = hi ? w3 : w1;

    v8f c = {};
    c = __builtin_amdgcn_wmma_f32_16x16x4_f32(
        /*neg_a=*/false, a, /*neg_b=*/false, bv,
        /*c_mod=*/(short)0, c, /*reuse_a=*/false, /*reuse_b=*/false);

    // lanes 0..7 write rows 0..7, lanes 16..23 write rows 8..15
    int l = lane & 15;
    if (l < 8) {
      int row = l + (hi ? 8 : 0);
      out[base + row] = c[l] + bb;
    }
  }

  // Scalar tail (nd is 16-divisible for this problem, but keep it general).
  int gidx    = (int)(blockIdx.x * blockDim.x + threadIdx.x);
  int gstride = (int)(gridDim.x * blockDim.x);
  for (int e = (ntiles << 4) + gidx; e < nd; e += gstride) {
    float s   = sum[e];
    float dg  = deg[e >> 5];
    bool  has = dg > 0.0f;
    float mnv = has ? mn[e] : 0.0f;
    float mxv = has ? mx[e] : 0.0f;
    float mean = s / fmaxf(dg, 1.0f);
    out[e] = w0 * s + w1 * mnv + w2 * mxv + w3 * mean + bb;
  }
}

// ---------------------------------------------------------------------------
extern "C" void kernel_launch(void* const* d_in, const int* in_sizes, int n_in,
                              void* d_out, int out_size, void* d_ws, size_t ws_size,
                              hipStream_t stream) {
  const float* m    = (const float*)d_in[0];
  const int*   dst  = (const int*)d_in[1];
  const float* w    = (const float*)d_in[2];
  const float* bptr = (const float*)d_in[3];

  const int n_edges = in_sizes[1];
  const int nd      = out_size;          // n_nodes * D_FEAT
  const int n_nodes = nd / D_FEAT;

  // Workspace layout: sum[nd] | min[nd] | max[nd] | deg[n_nodes]
  float* sum = (float*)d_ws;
  float* mn  = sum + nd;
  float* mx  = mn + nd;
  float* deg = mx + nd;

  {
    int threads = 256;
    int blocks  = (nd + threads - 1) / threads;
    seg_init_kernel<<<blocks, threads, 0, stream>>>(sum, mn, mx, deg, nd, n_nodes);
  }
  {
    long total  = (long)n_edges * 8;
    int threads = 256;
    int blocks  = (int)((total + threads - 1) / threads);
    seg_edge_kernel<<<blocks, threads, 0, stream>>>(m, dst, sum, mn, mx, deg, n_edges);
  }
  {
    int threads = 256;               // 8 waves per block
    int blocks  = 2048;              // grid-stride over 100K tiles
    seg_finalize_wmma_kernel<<<blocks, threads, 0, stream>>>(
        sum, mn, mx, deg, w, bptr, (float*)d_out, nd);
  }
}